// RotationalQuantizer_62277025792085
// MI455X (gfx1250) — compile-verified
//
#include <hip/hip_runtime.h>

#define B_ROWS 8192
#define D_DIM 64
#define C_CODES 2048
#define BETA 0.25f
#define EPS 1e-6f
#define RSQRT_D 0.125f   // 1/sqrt(64), exact

typedef __attribute__((ext_vector_type(2))) float v2f;
typedef __attribute__((ext_vector_type(8))) float v8f;

__device__ __forceinline__ float wave_sum32(float v) {
#pragma unroll
  for (int off = 16; off > 0; off >>= 1) v += __shfl_xor(v, off);
  return v;
}

// K0: squared L2 norm of each code row. one wave per code.
__global__ __launch_bounds__(256) void rq_code_norms(const float* __restrict__ codes,
                                                     float* __restrict__ cn) {
  int lane = threadIdx.x & 31;
  int wave = threadIdx.x >> 5;
  int code = blockIdx.x * 8 + wave;
  float2 c = *(const float2*)(codes + code * D_DIM + 2 * lane);
  float s = wave_sum32(c.x * c.x + c.y * c.y);
  if (lane == 0) cn[code] = s;
}

// K1: x_canonical = R^T x via closed-form plane rotation. one wave per row.
__global__ __launch_bounds__(256) void rq_rotate_to_canonical(const float* __restrict__ x,
                                                              const float* __restrict__ prev_q,
                                                              float* __restrict__ xc) {
  int lane = threadIdx.x & 31;
  int wave = threadIdx.x >> 5;
  int row = blockIdx.x * 8 + wave;
  float2 pq = *(const float2*)(prev_q + row * D_DIM + 2 * lane);
  float2 xv = *(const float2*)(x + row * D_DIM + 2 * lane);
  float s_pq2 = wave_sum32(pq.x * pq.x + pq.y * pq.y);
  float s_pq  = wave_sum32(pq.x + pq.y);
  float s_px  = wave_sum32(pq.x * xv.x + pq.y * xv.y);
  float s_x   = wave_sum32(xv.x + xv.y);
  float inv = 1.0f / fmaxf(sqrtf(s_pq2), EPS);
  float a  = s_px * inv;            // u . x
  float bv = RSQRT_D * s_x;         // v . x
  float c  = RSQRT_D * s_pq * inv;  // u . v
  float rdenom = 1.0f / (1.0f + c + EPS);
  float coefu = -bv + (c * bv - a) * rdenom;
  float coefv =  a  - (bv - c * a) * rdenom;
  float2 o;
  o.x = xv.x + coefu * (pq.x * inv) + coefv * RSQRT_D;
  o.y = xv.y + coefu * (pq.y * inv) + coefv * RSQRT_D;
  *(float2*)(xc + row * D_DIM + 2 * lane) = o;
}

// K2: fused dist-GEMM (f32 WMMA 16x16x4, K=64) + argmin over C=2048 codes.
// block = 8 waves = one 16-row M-tile; wave w scans N-tiles [w*16, w*16+16).
// B fragments double-buffered so next tile's 16 loads overlap this tile's WMMA chain.
__global__ __launch_bounds__(256) void rq_argmin_wmma(const float* __restrict__ xc,
                                                      const float* __restrict__ codes,
                                                      const float* __restrict__ cn,
                                                      float* __restrict__ out_idx) {
  __shared__ float sv[8][16];
  __shared__ int   si[8][16];
  int lane = threadIdx.x & 31;
  int wave = threadIdx.x >> 5;
  int m = lane & 15;
  int half = lane >> 4;
  int rowbase = blockIdx.x * 16;
  int nlane0 = wave * 256 + m;                 // this lane's code index at t=0

  // A fragments (f32 16x4): lane holds M=m, K = 4*kt + 2*half + {0,1}  (contiguous pair)
  v2f afrag[16];
  const float* ap = xc + (rowbase + m) * D_DIM + 2 * half;
#pragma unroll
  for (int kt = 0; kt < 16; ++kt) afrag[kt] = *(const v2f*)(ap + 4 * kt);

  // preload all 16 code-norm values for this wave's tiles
  float cnv[16];
#pragma unroll
  for (int t = 0; t < 16; ++t) cnv[t] = cn[nlane0 + t * 16];

  // prime double buffer with tile t=0's B fragments
  v2f bfr[2][16];
  {
    const float* cb = codes + nlane0 * D_DIM + 2 * half;
#pragma unroll
    for (int kt = 0; kt < 16; ++kt) bfr[0][kt] = *(const v2f*)(cb + 4 * kt);
  }

  float runv[8];
  int   runi[8];
#pragma unroll
  for (int r = 0; r < 8; ++r) { runv[r] = 3.4e38f; runi[r] = 0; }

#pragma unroll
  for (int t = 0; t < 16; ++t) {
    const int cur = t & 1, nxt = cur ^ 1;
    int n = nlane0 + t * 16;
    if (t < 15) {                              // issue next tile's loads before consuming current
      const float* cb = codes + (n + 16) * D_DIM + 2 * half;
      __builtin_prefetch(cb + 16 * D_DIM, 0, 1);   // tile t+2 toward L2
#pragma unroll
      for (int kt = 0; kt < 16; ++kt) bfr[nxt][kt] = *(const v2f*)(cb + 4 * kt);
    }
    v8f acc = {0.f, 0.f, 0.f, 0.f, 0.f, 0.f, 0.f, 0.f};
#pragma unroll
    for (int kt = 0; kt < 16; ++kt) {
      acc = __builtin_amdgcn_wmma_f32_16x16x4_f32(false, afrag[kt], false, bfr[cur][kt],
                                                  (short)0, acc, false, false);
    }
#pragma unroll
    for (int r = 0; r < 8; ++r) {              // C layout: vgpr r -> row r + 8*half
      float dist = cnv[t] - 2.0f * acc[r];
      if (dist < runv[r]) { runv[r] = dist; runi[r] = n; }
    }
  }

  // reduce over the 16 N-lanes of each half (tie -> lowest index, matching jnp.argmin)
#pragma unroll
  for (int off = 8; off > 0; off >>= 1) {
#pragma unroll
    for (int r = 0; r < 8; ++r) {
      float ov = __shfl_xor(runv[r], off);
      int   oi = __shfl_xor(runi[r], off);
      if (ov < runv[r] || (ov == runv[r] && oi < runi[r])) { runv[r] = ov; runi[r] = oi; }
    }
  }
  if (m == 0) {
#pragma unroll
    for (int r = 0; r < 8; ++r) { sv[wave][half * 8 + r] = runv[r]; si[wave][half * 8 + r] = runi[r]; }
  }
  __syncthreads();
  if (threadIdx.x < 16) {
    int row = threadIdx.x;
    float bv = sv[0][row];
    int   bi = si[0][row];
#pragma unroll
    for (int w = 1; w < 8; ++w) {              // ascending wave = ascending code range
      float v = sv[w][row];
      int   i = si[w][row];
      if (v < bv || (v == bv && i < bi)) { bv = v; bi = i; }
    }
    out_idx[rowbase + row] = (float)bi;
  }
}

// K3: quantized = R * codes[idx] (closed form) + per-row ||x - codes[idx]||^2.
__global__ __launch_bounds__(256) void rq_apply_rotation(const float* __restrict__ x,
                                                         const float* __restrict__ prev_q,
                                                         const float* __restrict__ codes,
                                                         const float* __restrict__ idx_f,
                                                         float* __restrict__ quant,
                                                         float* __restrict__ rowloss) {
  int lane = threadIdx.x & 31;
  int wave = threadIdx.x >> 5;
  int row = blockIdx.x * 8 + wave;
  int idx = (int)idx_f[row];
  float2 qc = *(const float2*)(codes + idx * D_DIM + 2 * lane);
  float2 pq = *(const float2*)(prev_q + row * D_DIM + 2 * lane);
  float2 xv = *(const float2*)(x + row * D_DIM + 2 * lane);
  float s_pq2 = wave_sum32(pq.x * pq.x + pq.y * pq.y);
  float s_pq  = wave_sum32(pq.x + pq.y);
  float s_pqc = wave_sum32(pq.x * qc.x + pq.y * qc.y);
  float s_qc  = wave_sum32(qc.x + qc.y);
  float d0 = xv.x - qc.x, d1 = xv.y - qc.y;
  float s_df = wave_sum32(d0 * d0 + d1 * d1);
  float inv = 1.0f / fmaxf(sqrtf(s_pq2), EPS);
  float aq = s_pqc * inv;            // u . q
  float bq = RSQRT_D * s_qc;         // v . q
  float c  = RSQRT_D * s_pq * inv;   // u . v
  float rdenom = 1.0f / (1.0f + c + EPS);
  float coefu =  bq + (c * bq - aq) * rdenom;
  float coefv = -aq - (bq - c * aq) * rdenom;
  float2 o;
  o.x = qc.x + coefu * (pq.x * inv) + coefv * RSQRT_D;
  o.y = qc.y + coefu * (pq.y * inv) + coefv * RSQRT_D;
  *(float2*)(quant + row * D_DIM + 2 * lane) = o;
  if (lane == 0) rowloss[row] = s_df;
}

// K4: deterministic fixed-order loss reduction (no float atomics).
__global__ __launch_bounds__(256) void rq_loss_reduce(const float* __restrict__ rowloss,
                                                      float* __restrict__ out_loss) {
  __shared__ float s[256];
  float acc = 0.f;
  for (int i = threadIdx.x; i < B_ROWS; i += 256) acc += rowloss[i];
  s[threadIdx.x] = acc;
  __syncthreads();
  for (int off = 128; off > 0; off >>= 1) {
    if ((int)threadIdx.x < off) s[threadIdx.x] += s[threadIdx.x + off];
    __syncthreads();
  }
  if (threadIdx.x == 0) *out_loss = (1.0f + BETA) * s[0] / (float)B_ROWS;
}

extern "C" void kernel_launch(void* const* d_in, const int* in_sizes, int n_in,
                              void* d_out, int out_size, void* d_ws, size_t ws_size,
                              hipStream_t stream) {
  (void)in_sizes; (void)n_in; (void)out_size; (void)ws_size;
  const float* x      = (const float*)d_in[0];
  const float* prev_q = (const float*)d_in[1];
  const float* codes  = (const float*)d_in[2];

  float* out      = (float*)d_out;
  float* xc       = out;                       // stage x_canonical in quantized slot (overwritten by K3)
  float* out_idx  = out + B_ROWS * D_DIM;      // indices as float
  float* out_loss = out + B_ROWS * D_DIM + B_ROWS;

  float* cn      = (float*)d_ws;               // C_CODES floats
  float* rowloss = cn + C_CODES;               // B_ROWS floats

  rq_code_norms<<<C_CODES / 8, 256, 0, stream>>>(codes, cn);
  rq_rotate_to_canonical<<<B_ROWS / 8, 256, 0, stream>>>(x, prev_q, xc);
  rq_argmin_wmma<<<B_ROWS / 16, 256, 0, stream>>>(xc, codes, cn, out_idx);
  rq_apply_rotation<<<B_ROWS / 8, 256, 0, stream>>>(x, prev_q, codes, out_idx, out, rowloss);
  rq_loss_reduce<<<1, 256, 0, stream>>>(rowloss, out_loss);
}